// SinkhornPermuter_85813446574743
// MI455X (gfx1250) — compile-verified
//
#include <hip/hip_runtime.h>

#define B_ 256
#define N_ 128
#define D_ 256
#define P_ 64
#define EPS_ 1e-4f
#define ITERS_ 100

typedef __attribute__((ext_vector_type(2))) float v2f;
typedef __attribute__((ext_vector_type(8))) float v8f;

__device__ __forceinline__ float wave_max(float v) {
#pragma unroll
    for (int o = 16; o > 0; o >>= 1) v = fmaxf(v, __shfl_xor(v, o, 32));
    return v;
}
__device__ __forceinline__ float wave_sum(float v) {
#pragma unroll
    for (int o = 16; o > 0; o >>= 1) v += __shfl_xor(v, o, 32);
    return v;
}

__global__ void sinkhorn_init_err(float* __restrict__ out) {
    if (threadIdx.x == 0 && blockIdx.x == 0)
        out[(size_t)B_ * N_ * N_] = 0.0f;
}

// One workgroup (8 wave32) per batch. Everything LDS-resident.
__global__ __launch_bounds__(256)
void sinkhorn_kernel(const float* __restrict__ embIn,
                     const unsigned char* __restrict__ maskIn,
                     const float* __restrict__ embOut,
                     const float* __restrict__ padv,
                     const float* __restrict__ posEnc,
                     const float* __restrict__ Win,
                     const float* __restrict__ bIn,
                     const float* __restrict__ Wout,
                     const float* __restrict__ bOut,
                     float* __restrict__ out)
{
    __shared__ float sA[N_][P_ + 1];   // a = proj(outputs+pos)   (i index)
    __shared__ float sB[N_][P_ + 1];   // b = proj(masked inputs) (j index)
    __shared__ float sC[N_][N_ + 1];   // C, then log_K, then log_P
    __shared__ float sRed[8];
    __shared__ float sScale;

    const int bb   = blockIdx.x;
    const int tid  = threadIdx.x;
    const int wave = tid >> 5;
    const int lane = tid & 31;
    const int lr   = lane & 15;   // lane-in-half (M for A frag, N for B frag)
    const int half = lane >> 4;   // selects K pair {0,1} vs {2,3}

    const float* eOut = embOut + (size_t)bb * N_ * D_;
    const float* eIn  = embIn  + (size_t)bb * N_ * D_;
    const unsigned char* msk = maskIn + (size_t)bb * N_;

    // ---------------- Stage 1: two GEMMs via V_WMMA_F32_16X16X4_F32 --------
    // wave w owns output rows [w*16, w*16+16), all 4 col tiles of P=64.
    {
        const int row  = wave * 16 + lr;              // A-frag M index
        const bool m   = msk[row] != 0;
        const v8f zero = {0.f, 0.f, 0.f, 0.f, 0.f, 0.f, 0.f, 0.f};
        v8f accA[4], accB[4];
#pragma unroll
        for (int c = 0; c < 4; ++c) { accA[c] = zero; accB[c] = zero; }

        for (int k0 = 0; k0 < D_; k0 += 4) {
            const int kk = k0 + 2 * half;
            // A fragments (16x4 fp32): lane holds X[row][kk], X[row][kk+1]
            v2f aO, aI;
            aO.x = eOut[row * D_ + kk]     + posEnc[row * D_ + kk];
            aO.y = eOut[row * D_ + kk + 1] + posEnc[row * D_ + kk + 1];
            float i0 = eIn[row * D_ + kk];
            float i1 = eIn[row * D_ + kk + 1];
            aI.x = m ? padv[kk]     : i0;
            aI.y = m ? padv[kk + 1] : i1;
#pragma unroll
            for (int c = 0; c < 4; ++c) {
                const int col = c * 16 + lr;          // B-frag N index
                v2f bO, bI;
                bO.x = Wout[kk * P_ + col]; bO.y = Wout[(kk + 1) * P_ + col];
                bI.x = Win [kk * P_ + col]; bI.y = Win [(kk + 1) * P_ + col];
                accA[c] = __builtin_amdgcn_wmma_f32_16x16x4_f32(
                    false, aO, false, bO, (short)0, accA[c], false, false);
                accB[c] = __builtin_amdgcn_wmma_f32_16x16x4_f32(
                    false, aI, false, bI, (short)0, accB[c], false, false);
            }
        }
        // C/D layout: vgpr r, lanes<16 -> M=r, lanes>=16 -> M=r+8; N = lane&15
#pragma unroll
        for (int c = 0; c < 4; ++c) {
            const int col = c * 16 + lr;
#pragma unroll
            for (int r = 0; r < 8; ++r) {
                const int ro = wave * 16 + r + 8 * half;
                sA[ro][col] = accA[c][r] + bOut[col];
                sB[ro][col] = accB[c][r] + bIn[col];
            }
        }
    }
    __syncthreads();

    // ---------------- Stage 2: pairwise L1 cost + global sum ----------------
    float partial = 0.f;
    for (int e = tid; e < N_ * N_; e += 256) {
        const int i = e >> 7, j = e & 127;
        float s = 0.f;
#pragma unroll 8
        for (int k = 0; k < P_; ++k) s += fabsf(sA[i][k] - sB[j][k]);
        sC[i][j] = s;
        partial += s;
    }
    partial = wave_sum(partial);
    if (lane == 0) sRed[wave] = partial;
    __syncthreads();
    if (tid == 0) {
        float t = 0.f;
        for (int w = 0; w < 8; ++w) t += sRed[w];
        sScale = -1.0f / (t * EPS_);   // log_K = -(C/sum)/eps
    }
    __syncthreads();
    {
        const float scale = sScale;
        for (int e = tid; e < N_ * N_; e += 256) {
            const int i = e >> 7, j = e & 127;
            sC[i][j] *= scale;
        }
    }
    __syncthreads();

    // ---------------- Stage 3: 100 Sinkhorn iterations in LDS ---------------
    for (int it = 0; it < ITERS_; ++it) {
        // rows: subtract logsumexp over j (one wave per row)
        for (int r = wave; r < N_; r += 8) {
            float v0 = sC[r][lane],      v1 = sC[r][lane + 32];
            float v2 = sC[r][lane + 64], v3 = sC[r][lane + 96];
            float mx = wave_max(fmaxf(fmaxf(v0, v1), fmaxf(v2, v3)));
            float s  = __expf(v0 - mx) + __expf(v1 - mx) +
                       __expf(v2 - mx) + __expf(v3 - mx);
            s = wave_sum(s);
            const float lse = mx + __logf(s);
            sC[r][lane]      = v0 - lse; sC[r][lane + 32] = v1 - lse;
            sC[r][lane + 64] = v2 - lse; sC[r][lane + 96] = v3 - lse;
        }
        __syncthreads();
        // cols: subtract logsumexp over i (one wave per column; stride 129 keeps banks distinct)
        for (int c = wave; c < N_; c += 8) {
            float v0 = sC[lane][c],      v1 = sC[lane + 32][c];
            float v2 = sC[lane + 64][c], v3 = sC[lane + 96][c];
            float mx = wave_max(fmaxf(fmaxf(v0, v1), fmaxf(v2, v3)));
            float s  = __expf(v0 - mx) + __expf(v1 - mx) +
                       __expf(v2 - mx) + __expf(v3 - mx);
            s = wave_sum(s);
            const float lse = mx + __logf(s);
            sC[lane][c]      = v0 - lse; sC[lane + 32][c] = v1 - lse;
            sC[lane + 64][c] = v2 - lse; sC[lane + 96][c] = v3 - lse;
        }
        __syncthreads();
    }

    // ---------------- Stage 4: P = exp(log_P), write out, err --------------
    float* outB = out + (size_t)bb * N_ * N_;
    float errmax = 0.f;
    for (int r = wave; r < N_; r += 8) {
        float v0 = __expf(sC[r][lane]);
        float v1 = __expf(sC[r][lane + 32]);
        float v2 = __expf(sC[r][lane + 64]);
        float v3 = __expf(sC[r][lane + 96]);
        outB[r * N_ + lane]      = v0;
        outB[r * N_ + lane + 32] = v1;
        outB[r * N_ + lane + 64] = v2;
        outB[r * N_ + lane + 96] = v3;
        const float rs = wave_sum(v0 + v1 + v2 + v3);
        errmax = fmaxf(errmax, fabsf(rs - 1.0f));
    }
    errmax = wave_max(errmax);
    if (lane == 0) sRed[wave] = errmax;
    __syncthreads();
    if (tid == 0) {
        float t = 0.f;
        for (int w = 0; w < 8; ++w) t = fmaxf(t, sRed[w]);
        // err >= 0 so uint bit-pattern ordering matches float ordering
        atomicMax((unsigned int*)(out + (size_t)B_ * N_ * N_),
                  __float_as_uint(t));
    }
}

extern "C" void kernel_launch(void* const* d_in, const int* in_sizes, int n_in,
                              void* d_out, int out_size, void* d_ws, size_t ws_size,
                              hipStream_t stream) {
    (void)in_sizes; (void)n_in; (void)out_size; (void)d_ws; (void)ws_size;
    const float*         embIn  = (const float*)d_in[0];
    const unsigned char* mask   = (const unsigned char*)d_in[1];  // jnp bool -> 1 byte
    const float*         embOut = (const float*)d_in[2];
    const float*         padv   = (const float*)d_in[3];
    const float*         pos    = (const float*)d_in[4];
    const float*         Win    = (const float*)d_in[5];
    const float*         bIn    = (const float*)d_in[6];
    const float*         Wout   = (const float*)d_in[7];
    const float*         bOut   = (const float*)d_in[8];
    float* out = (float*)d_out;

    hipLaunchKernelGGL(sinkhorn_init_err, dim3(1), dim3(1), 0, stream, out);
    hipLaunchKernelGGL(sinkhorn_kernel, dim3(B_), dim3(256), 0, stream,
                       embIn, mask, embOut, padv, pos, Win, bIn, Wout, bOut, out);
}